// FeatureEmbeddinng_58394375357022
// MI455X (gfx1250) — compile-verified
//
#include <hip/hip_runtime.h>

typedef float v2f __attribute__((ext_vector_type(2)));
typedef float v8f __attribute__((ext_vector_type(8)));

#define H      128
#define KTXN   371
#define VOCAB  1000

// ---------------------------------------------------------------------------
// Categorical path: one wave per node row; each lane copies a float4.
// 32 lanes x 16B = 128 floats = one embedding row. Tables are 2MB -> L2-hot.
// ---------------------------------------------------------------------------
__global__ void __launch_bounds__(256) cat_gather_kernel(
    const float* __restrict__ tables,
    const int*   __restrict__ type_ids,
    const int*   __restrict__ value_ids,
    float*       __restrict__ out,
    int n_cat) {
  int t    = blockIdx.x * blockDim.x + threadIdx.x;
  int row  = t >> 5;
  int lane = t & 31;
  if (row >= n_cat) return;
  int ty = type_ids[row];
  int va = value_ids[row];
  const float4* src = (const float4*)(tables + ((size_t)ty * VOCAB + (size_t)va) * H);
  float4*       dst = (float4*)(out + (size_t)row * H);
  dst[lane] = src[lane];
}

// ---------------------------------------------------------------------------
// Continuous path: one wave per node row; out = v * w[type] + b[type].
// w/b are 4x128 floats (2KB each) -> cache-resident.
// ---------------------------------------------------------------------------
__global__ void __launch_bounds__(256) cont_affine_kernel(
    const float* __restrict__ w,
    const float* __restrict__ b,
    const float* __restrict__ vals,
    const int*   __restrict__ type_ids,
    float*       __restrict__ out,
    int n_cont) {
  int t    = blockIdx.x * blockDim.x + threadIdx.x;
  int row  = t >> 5;
  int lane = t & 31;
  if (row >= n_cont) return;
  int   ty = type_ids[row];
  float v  = vals[row];
  const float4* w4 = (const float4*)(w + (size_t)ty * H);
  const float4* b4 = (const float4*)(b + (size_t)ty * H);
  float4 ww = w4[lane];
  float4 bb = b4[lane];
  float4 o;
  o.x = fmaf(v, ww.x, bb.x);
  o.y = fmaf(v, ww.y, bb.y);
  o.z = fmaf(v, ww.z, bb.z);
  o.w = fmaf(v, ww.w, bb.w);
  ((float4*)(out + (size_t)row * H))[lane] = o;
}

// ---------------------------------------------------------------------------
// Transaction path: fp32 WMMA GEMM  [M x 371] @ [371 x 128] + bias.
// One wave per 16-row strip, 8 x (16x16) accumulator tiles covering all 128
// output columns. K loop: 92 full 16x16x4 steps + 1 zero-padded tail step.
// A fragment (v2f): lane L holds A[m0 + (L&15)][k0 + 2*(L>>4) + {0,1}].
// B fragment (v2f): lane L holds B[k0 + 2*(L>>4) + {0,1}][n0 + (L&15)].
// C/D tile  (v8f): lane L, vgpr r holds C[m0 + r + 8*(L>>4)][n0 + (L&15)].
// ---------------------------------------------------------------------------
__global__ void __launch_bounds__(256, 2) txn_gemm_kernel(
    const float* __restrict__ feats,   // [M x 371] row-major
    const float* __restrict__ w,       // [371 x 128] row-major
    const float* __restrict__ bias,    // [128]
    float*       __restrict__ out,     // [M x 128]
    int n_txn) {
  const int lane  = threadIdx.x & 31;
  const int wave  = threadIdx.x >> 5;
  const int strip = blockIdx.x * 8 + wave;   // wave-uniform
  const int m0    = strip * 16;
  if (m0 >= n_txn) return;                    // uniform branch: EXEC stays all-1s

  const int half = lane >> 4;                 // 0 or 1: which K pair this lane holds
  const int col  = lane & 15;                 // fixed column / A row within strip

  const float* arow = feats + (size_t)(m0 + col) * KTXN;

  v8f acc[8];
#pragma unroll
  for (int j = 0; j < 8; ++j) acc[j] = (v8f){0.f, 0.f, 0.f, 0.f, 0.f, 0.f, 0.f, 0.f};

  constexpr int KFULL = (KTXN / 4) * 4;       // 368

  for (int k0 = 0; k0 < KFULL; k0 += 4) {
    const int ka = k0 + 2 * half;
    v2f a;
    a.x = arow[ka];
    a.y = arow[ka + 1];
    const float* wk = w + (size_t)ka * H + col;
#pragma unroll
    for (int j = 0; j < 8; ++j) {
      v2f bf;
      bf.x = wk[j * 16];
      bf.y = wk[H + j * 16];
      acc[j] = __builtin_amdgcn_wmma_f32_16x16x4_f32(
          /*neg_a=*/false, a, /*neg_b=*/false, bf,
          /*c_mod=*/(short)0, acc[j], /*reuse_a=*/false, /*reuse_b=*/false);
    }
  }

  { // K tail: k = 368..370 (371 is zero-padded on both A and B sides)
    const int  ka    = KFULL + 2 * half;      // 368 or 370
    const bool hasKb = (ka + 1) < KTXN;       // false only for lanes 16..31
    v2f a;
    a.x = arow[ka];
    a.y = hasKb ? arow[ka + 1] : 0.0f;
    const float* wk = w + (size_t)ka * H + col;
#pragma unroll
    for (int j = 0; j < 8; ++j) {
      v2f bf;
      bf.x = wk[j * 16];
      bf.y = hasKb ? wk[H + j * 16] : 0.0f;
      acc[j] = __builtin_amdgcn_wmma_f32_16x16x4_f32(
          false, a, false, bf, (short)0, acc[j], false, false);
    }
  }

  // Bias add + store. Lane's column is fixed (col); rows are r + 8*half.
#pragma unroll
  for (int j = 0; j < 8; ++j) {
    const float bj = bias[j * 16 + col];
#pragma unroll
    for (int r = 0; r < 8; ++r) {
      out[(size_t)(m0 + r + 8 * half) * H + j * 16 + col] = acc[j][r] + bj;
    }
  }
}

// ---------------------------------------------------------------------------
// Launcher
// ---------------------------------------------------------------------------
extern "C" void kernel_launch(void* const* d_in, const int* in_sizes, int n_in,
                              void* d_out, int out_size, void* d_ws, size_t ws_size,
                              hipStream_t stream) {
  (void)n_in; (void)out_size; (void)d_ws; (void)ws_size;

  const float* cat_tables   = (const float*)d_in[0];  // [4,1000,128]
  const float* cont_w       = (const float*)d_in[1];  // [4,128]
  const float* cont_b       = (const float*)d_in[2];  // [4,128]
  const float* txn_w        = (const float*)d_in[3];  // [371,128]
  const float* txn_b        = (const float*)d_in[4];  // [128]
  const float* txn_feats    = (const float*)d_in[5];  // [N_TXN,371]
  const float* cont_values  = (const float*)d_in[6];  // [N_CONT]
  const int*   cat_type_ids = (const int*)d_in[7];    // [N_CAT]
  const int*   cat_value_ids= (const int*)d_in[8];    // [N_CAT]
  const int*   cont_type_ids= (const int*)d_in[9];    // [N_CONT]

  const int n_cat  = in_sizes[7];
  const int n_cont = in_sizes[6];
  const int n_txn  = in_sizes[5] / KTXN;

  float* out_cat  = (float*)d_out;
  float* out_cont = out_cat  + (size_t)n_cat  * H;
  float* out_txn  = out_cont + (size_t)n_cont * H;

  // Categorical gather: one wave per row.
  {
    long long threads = (long long)n_cat * 32;
    int blocks = (int)((threads + 255) / 256);
    cat_gather_kernel<<<blocks, 256, 0, stream>>>(
        cat_tables, cat_type_ids, cat_value_ids, out_cat, n_cat);
  }

  // Continuous affine: one wave per row.
  {
    long long threads = (long long)n_cont * 32;
    int blocks = (int)((threads + 255) / 256);
    cont_affine_kernel<<<blocks, 256, 0, stream>>>(
        cont_w, cont_b, cont_values, cont_type_ids, out_cont, n_cont);
  }

  // Transaction GEMM: 8 waves/block, each wave = one 16x128 strip.
  {
    int strips = (n_txn + 15) / 16;
    int blocks = (strips + 7) / 8;
    txn_gemm_kernel<<<blocks, 256, 0, stream>>>(
        txn_feats, txn_w, txn_b, out_txn, n_txn);
  }
}